// ContrastiveCrossLingualLearning_34024730919224
// MI455X (gfx1250) — compile-verified
//
#include <hip/hip_runtime.h>
#include <math.h>

typedef float v2f __attribute__((ext_vector_type(2)));
typedef float v4f __attribute__((ext_vector_type(4)));
typedef float v8f __attribute__((ext_vector_type(8)));

#define B_ 128
#define S_ 2048
#define H_ 1024
#define C_ 256

// workspace layout (float offsets)
#define WS_POOLED 0
#define WS_HBUF   131072
#define WS_PROJ   262144
#define WS_SIM    294912
#define WS_COUNTS 311296
#define WS_LOSS   311424

__global__ void zero_kernel(float* __restrict__ p, int n) {
  int i = blockIdx.x * blockDim.x + threadIdx.x;
  if (i < n) p[i] = 0.f;
}

// ---------------------------------------------------------------------------
// Masked mean pool, phase 1: partial sums. 1.07 GB streaming read => dominant
// cost; use 128-bit non-temporal loads (doesn't fit 192MB L2) + f32 atomics.
// grid = B*8 (8 S-chunks per batch row), block = 256 threads * 4 floats = H.
// ---------------------------------------------------------------------------
__global__ __launch_bounds__(256) void pool_partial_kernel(
    const float* __restrict__ feat, const int* __restrict__ mask,
    float* __restrict__ pooled, float* __restrict__ counts) {
  const int SCHUNK = S_ / 8;
  int b  = blockIdx.x >> 3;
  int s0 = (blockIdx.x & 7) * SCHUNK;
  int t  = threadIdx.x;
  v4f  acc = {0.f, 0.f, 0.f, 0.f};
  float cnt = 0.f;
  const float* fb = feat + (size_t)b * S_ * H_;
  for (int s = 0; s < SCHUNK; ++s) {
    int srow = s0 + s;
    float m = (float)mask[b * S_ + srow];
    const v4f* p = (const v4f*)(fb + (size_t)srow * H_) + t;
    if (s + 8 < SCHUNK)
      __builtin_prefetch(fb + (size_t)(srow + 8) * H_ + t * 4, 0, 0);
    v4f v = __builtin_nontemporal_load(p);      // TH=NT streaming read
    acc.x += v.x * m; acc.y += v.y * m; acc.z += v.z * m; acc.w += v.w * m;
    cnt += m;
  }
  float* pd = pooled + b * H_ + t * 4;
  atomicAdd(pd + 0, acc.x);
  atomicAdd(pd + 1, acc.y);
  atomicAdd(pd + 2, acc.z);
  atomicAdd(pd + 3, acc.w);
  if (t == 0) atomicAdd(counts + b, cnt);
}

__global__ void pool_divide_kernel(float* __restrict__ pooled,
                                   const float* __restrict__ counts) {
  int i = blockIdx.x * blockDim.x + threadIdx.x;
  if (i < B_ * H_) pooled[i] /= counts[i / H_];
}

// ---------------------------------------------------------------------------
// f32 GEMM via V_WMMA_F32_16X16X4_F32. One 16x16 output tile per wave,
// 4 waves per 128-thread block. A: [M,K] row-major, B: [K,N] row-major.
// A-frag layout (ISA 7.12.2, 32-bit A 16x4): lanes 0-15 hold K=0,1 (VGPR0,1);
// lanes 16-31 hold K=2,3 -> contiguous float2 per lane.
// C/D layout: VGPR v -> row m0+v (lanes 0-15) / m0+v+8 (lanes 16-31), col=lane%16.
// ---------------------------------------------------------------------------
__global__ __launch_bounds__(128) void wmma_gemm_bias_kernel(
    const float* __restrict__ A, const float* __restrict__ Bm,
    const float* __restrict__ bias, float* __restrict__ Out,
    int M, int N, int K, int relu) {
  int wave = threadIdx.x >> 5;
  int lane = threadIdx.x & 31;
  int tile = blockIdx.x * 4 + wave;
  int ntn  = N >> 4;
  int m0   = (tile / ntn) << 4;
  int n0   = (tile % ntn) << 4;
  int half = lane >> 4;
  int l    = lane & 15;

  const float* ap = A  + (size_t)(m0 + l) * K + half * 2;
  const float* bp = Bm + (size_t)(n0 + l);
  v8f c = {};
  for (int k0 = 0; k0 < K; k0 += 4) {
    v2f a = *(const v2f*)(ap + k0);              // contiguous b64 load
    int kb = k0 + half * 2;
    v2f b;
    b.x = bp[(size_t)kb * N];
    b.y = bp[(size_t)(kb + 1) * N];
    c = __builtin_amdgcn_wmma_f32_16x16x4_f32(false, a, false, b,
                                              (short)0, c, false, false);
  }
  int col = n0 + l;
  float bv = bias[col];
#pragma unroll
  for (int v = 0; v < 8; ++v) {
    int row = m0 + v + half * 8;
    float x = c[v] + bv;
    if (relu) x = fmaxf(x, 0.f);
    Out[(size_t)row * N + col] = x;
  }
}

// L2-normalize each 256-wide row of proj; writes normalized proj to output.
__global__ __launch_bounds__(256) void normalize_kernel(
    const float* __restrict__ proj, float* __restrict__ out) {
  __shared__ float red[256];
  int b = blockIdx.x, t = threadIdx.x;
  float x = proj[b * C_ + t];
  red[t] = x * x;
  __syncthreads();
  for (int s = 128; s > 0; s >>= 1) {
    if (t < s) red[t] += red[t + s];
    __syncthreads();
  }
  float norm = fmaxf(sqrtf(red[0]), 1e-12f);
  out[b * C_ + t] = x / norm;
}

// sim = (P P^T)/temp via WMMA; both A and B fragments are contiguous float2
// loads since B[k][n] = P[n][k] and P is row-major [128,256].
__global__ __launch_bounds__(128) void wmma_sim_kernel(
    const float* __restrict__ P, const float* __restrict__ temp,
    float* __restrict__ Sim) {
  const int K = C_, N = B_;
  int wave = threadIdx.x >> 5;
  int lane = threadIdx.x & 31;
  int tile = blockIdx.x * 4 + wave;
  int ntn  = N >> 4;
  int m0   = (tile / ntn) << 4;
  int n0   = (tile % ntn) << 4;
  int half = lane >> 4;
  int l    = lane & 15;

  const float* ap = P + (size_t)(m0 + l) * K + half * 2;
  const float* bp = P + (size_t)(n0 + l) * K + half * 2;
  v8f c = {};
  for (int k0 = 0; k0 < K; k0 += 4) {
    v2f a = *(const v2f*)(ap + k0);
    v2f b = *(const v2f*)(bp + k0);
    c = __builtin_amdgcn_wmma_f32_16x16x4_f32(false, a, false, b,
                                              (short)0, c, false, false);
  }
  float inv_t = 1.f / temp[0];
  int col = n0 + l;
#pragma unroll
  for (int v = 0; v < 8; ++v) {
    int row = m0 + v + half * 8;
    Sim[row * N + col] = c[v] * inv_t;
  }
}

// Per-row soft-label cross entropy: rowloss = sum_j lab_ij*(LSE_i - sim_ij).
__global__ __launch_bounds__(128) void loss_kernel(
    const float* __restrict__ Sim, const int* __restrict__ lang,
    float* __restrict__ lossacc) {
  __shared__ float red[128];
  int i = blockIdx.x, j = threadIdx.x;
  float s = Sim[i * B_ + j];

  red[j] = s;                                  // row max
  __syncthreads();
  for (int st = 64; st > 0; st >>= 1) {
    if (j < st) red[j] = fmaxf(red[j], red[j + st]);
    __syncthreads();
  }
  float smax = red[0];
  __syncthreads();

  red[j] = expf(s - smax);                     // sum exp
  __syncthreads();
  for (int st = 64; st > 0; st >>= 1) {
    if (j < st) red[j] += red[j + st];
    __syncthreads();
  }
  float lse = smax + logf(red[0]);
  __syncthreads();

  float lab = ((lang[i] != lang[j]) && (i != j)) ? 1.f : 0.f;
  red[j] = lab * (lse - s);                    // weighted NLL
  __syncthreads();
  for (int st = 64; st > 0; st >>= 1) {
    if (j < st) red[j] += red[j + st];
    __syncthreads();
  }
  if (j == 0) atomicAdd(lossacc, red[0] * (1.f / (float)B_));
}

__global__ void finalize_kernel(const float* __restrict__ lossacc,
                                float* __restrict__ out_loss) {
  if (threadIdx.x == 0 && blockIdx.x == 0) out_loss[0] = lossacc[0];
}

extern "C" void kernel_launch(void* const* d_in, const int* in_sizes, int n_in,
                              void* d_out, int out_size, void* d_ws, size_t ws_size,
                              hipStream_t stream) {
  const float* features = (const float*)d_in[0];
  const float* W1       = (const float*)d_in[1];
  const float* b1       = (const float*)d_in[2];
  const float* W2       = (const float*)d_in[3];
  const float* b2       = (const float*)d_in[4];
  const float* temp     = (const float*)d_in[5];
  const int*   lang     = (const int*)d_in[6];
  const int*   amask    = (const int*)d_in[7];

  float* ws      = (float*)d_ws;
  float* pooled  = ws + WS_POOLED;   // [B,H]
  float* hbuf    = ws + WS_HBUF;     // [B,H]
  float* proj    = ws + WS_PROJ;     // [B,C] pre-norm
  float* simbuf  = ws + WS_SIM;      // [B,B]
  float* counts  = ws + WS_COUNTS;   // [B]
  float* lossacc = ws + WS_LOSS;     // [1]

  float* out_proj = (float*)d_out;            // [B,C] normalized
  float* out_loss = out_proj + B_ * C_;       // scalar

  // deterministic: re-zero accumulators every launch
  zero_kernel<<<(B_ * H_ + 255) / 256, 256, 0, stream>>>(pooled, B_ * H_);
  zero_kernel<<<1, 256, 0, stream>>>(counts, B_ + 1);  // counts + lossacc (adjacent)

  // 1) masked mean pool (HBM-bound: ~1.07 GB streaming read)
  pool_partial_kernel<<<B_ * 8, 256, 0, stream>>>(features, amask, pooled, counts);
  pool_divide_kernel<<<(B_ * H_ + 255) / 256, 256, 0, stream>>>(pooled, counts);

  // 2) MLP head via f32 WMMA
  wmma_gemm_bias_kernel<<<(8 * 64) / 4, 128, 0, stream>>>(
      pooled, W1, b1, hbuf, B_, H_, H_, 1);          // relu(pooled@W1+b1)
  wmma_gemm_bias_kernel<<<(8 * 16) / 4, 128, 0, stream>>>(
      hbuf, W2, b2, proj, B_, C_, H_, 0);            // h@W2+b2

  // 3) L2 normalize -> d_out proj region
  normalize_kernel<<<B_, 256, 0, stream>>>(proj, out_proj);

  // 4) similarity matrix via WMMA
  wmma_sim_kernel<<<(8 * 8) / 4, 128, 0, stream>>>(out_proj, temp, simbuf);

  // 5) soft-label cross entropy, mean over rows
  loss_kernel<<<B_, 128, 0, stream>>>(simbuf, lang, lossacc);
  finalize_kernel<<<1, 1, 0, stream>>>(lossacc, out_loss);
}